// GATV2_24481313587664
// MI455X (gfx1250) — compile-verified
//
#include <hip/hip_runtime.h>
#include <hip/hip_bf16.h>
#include <math.h>

typedef float v2f __attribute__((ext_vector_type(2)));
typedef float v8f __attribute__((ext_vector_type(8)));

namespace cfg {
constexpr int IN_C = 512, HID = 128, OUT_C = 40, H = 8, C = 16, L = 5;
}

// ---------------------------------------------------------------------------
// init kernels
// ---------------------------------------------------------------------------
__global__ void zero2_kernel(float* __restrict__ a, float* __restrict__ b, int n) {
    int i = blockIdx.x * blockDim.x + threadIdx.x;
    if (i < n) { a[i] = 0.f; b[i] = 0.f; }
}

__global__ void init_layer_kernel(float* __restrict__ agg, float* __restrict__ emax,
                                  float* __restrict__ denom, int nHID, int nH) {
    int i = blockIdx.x * blockDim.x + threadIdx.x;
    if (i < nHID) agg[i] = 0.f;
    if (i < nH) { emax[i] = -INFINITY; denom[i] = 0.f; }
}

// ---------------------------------------------------------------------------
// fp32 WMMA GEMM: C[M,NCOL] = A[M,K] * W[K,NCOL] (+ bias)
// One wave computes TN adjacent 16x16 tiles (shared A fragment) via
// V_WMMA_F32_16X16X4_F32.  Fragment layouts (CDNA5 ISA 7.12.2, 32-bit):
//   A 16x4 : lane l -> row = l%16 ; elem j -> k = 2*(l/16) + j
//   B 4x16 : lane l -> col = l%16 ; elem j -> k = 2*(l/16) + j
//   C/D    : lane l -> col = l%16 ; elem j -> row = j + 8*(l/16)
// MASK=false assumes M%16==0 and all tile columns < NCOL (no cndmasks in the
// hot loop); MASK=true bounds-checks rows and columns (fc1, NCOL=40).
// ---------------------------------------------------------------------------
template<int K, int NCOL, int TN, bool MASK, bool ADD_BIAS>
__global__ void gemm_wmma_f32(const float* __restrict__ A, const float* __restrict__ W,
                              const float* __restrict__ bias, float* __restrict__ Cmat,
                              int M) {
    const int lane = threadIdx.x & 31;
    const int wave = threadIdx.x >> 5;
    const int m0   = blockIdx.x * 16;
    const int n0   = wave * 16 * TN;
    const int half = lane >> 4;   // 0 or 1
    const int l16  = lane & 15;
    const int arow = m0 + l16;

    bool avalid = true;
    if (MASK) avalid = (arow < M);

    const float* Ap = A + (size_t)(MASK ? (avalid ? arow : 0) : arow) * K + half * 2;

    int  bcol[TN];
    bool bvalid[TN];
    const float* Wp[TN];
#pragma unroll
    for (int t = 0; t < TN; ++t) {
        bcol[t]   = n0 + t * 16 + l16;
        bvalid[t] = MASK ? (bcol[t] < NCOL) : true;
        Wp[t] = W + (size_t)(half * 2) * NCOL + (bvalid[t] ? bcol[t] : 0);
    }

    v8f acc[TN];
#pragma unroll
    for (int t = 0; t < TN; ++t) acc[t] = v8f{};

#pragma unroll 4
    for (int kk = 0; kk < K; kk += 4) {
        v2f a = *(const v2f*)(Ap + kk);          // one b64 load, 8B aligned
        if (MASK && !avalid) { a.x = 0.f; a.y = 0.f; }
#pragma unroll
        for (int t = 0; t < TN; ++t) {
            v2f b;
            b.x = Wp[t][(size_t)kk * NCOL];
            b.y = Wp[t][(size_t)(kk + 1) * NCOL];
            if (MASK && !bvalid[t]) { b.x = 0.f; b.y = 0.f; }
            acc[t] = __builtin_amdgcn_wmma_f32_16x16x4_f32(
                /*neg_a=*/false, a, /*neg_b=*/false, b,
                /*c_mod=*/(short)0, acc[t], /*reuse_a=*/false, /*reuse_b=*/false);
        }
    }

#pragma unroll
    for (int t = 0; t < TN; ++t) {
        float bv = 0.f;
        if (ADD_BIAS && bvalid[t]) bv = bias[bcol[t]];
#pragma unroll
        for (int j = 0; j < 8; ++j) {
            int row = m0 + j + 8 * half;
            bool ok = MASK ? (bvalid[t] && row < M) : true;
            if (ok) Cmat[(size_t)row * NCOL + bcol[t]] = acc[t][j] + bv;
        }
    }
}

// ---------------------------------------------------------------------------
// per-(node,head) attention logits: a_src[n,h] = <hp[n,h,:], att_src[h,:]>
// ---------------------------------------------------------------------------
__global__ void att_kernel(const float* __restrict__ hp, const float* __restrict__ wsrc,
                           const float* __restrict__ wdst, float* __restrict__ a_s,
                           float* __restrict__ a_d, int NH) {
    using namespace cfg;
    int idx = blockIdx.x * blockDim.x + threadIdx.x;   // n*H + h
    if (idx >= NH) return;
    int h = idx & (H - 1);
    const float* v  = hp + (size_t)(idx >> 3) * HID + h * C;
    const float* ps = wsrc + h * C;
    const float* pd = wdst + h * C;
    float ss = 0.f, sd = 0.f;
#pragma unroll
    for (int c = 0; c < C; ++c) {
        float x = v[c];
        ss += x * ps[c];
        sd += x * pd[c];
    }
    a_s[idx] = ss;
    a_d[idx] = sd;
}

__device__ __forceinline__ void edge_sd(const int* __restrict__ ei, int e, int E,
                                        int& s, int& d) {
    if (e < E) { s = ei[e]; d = ei[E + e]; }
    else       { s = e - E; d = e - E; }     // appended self loops
}

__device__ __forceinline__ float leaky02(float x) { return x > 0.f ? x : 0.2f * x; }

// pass 1: segment max of leaky_relu(a_src[s]+a_dst[d]) per (dst,head)
__global__ void edge_max_kernel(const int* __restrict__ ei, const float* __restrict__ a_s,
                                const float* __restrict__ a_d, float* __restrict__ emax,
                                int E, int EH) {
    using namespace cfg;
    int idx = blockIdx.x * blockDim.x + threadIdx.x;   // e*H + h
    if (idx >= EH) return;
    int e = idx >> 3, h = idx & (H - 1);
    int s, d; edge_sd(ei, e, E, s, d);
    float v = leaky02(a_s[s * H + h] + a_d[d * H + h]);
    atomicMax(&emax[d * H + h], v);
}

// pass 2: p = exp(v - emax[dst]); store per-edge p; segment sum into denom
__global__ void edge_exp_sum_kernel(const int* __restrict__ ei, const float* __restrict__ a_s,
                                    const float* __restrict__ a_d, const float* __restrict__ emax,
                                    float* __restrict__ denom, float* __restrict__ pedge,
                                    int E, int EH) {
    using namespace cfg;
    int idx = blockIdx.x * blockDim.x + threadIdx.x;
    if (idx >= EH) return;
    int e = idx >> 3, h = idx & (H - 1);
    int s, d; edge_sd(ei, e, E, s, d);
    float v = leaky02(a_s[s * H + h] + a_d[d * H + h]);
    float p = expf(v - emax[d * H + h]);
    pedge[idx] = p;
    atomicAdd(&denom[d * H + h], p);
}

// pass 3: agg[d,h,:] += alpha * hp[s,h,:]
__global__ void edge_scatter_kernel(const int* __restrict__ ei, const float* __restrict__ hp,
                                    const float* __restrict__ pedge, const float* __restrict__ denom,
                                    float* __restrict__ agg, int E, int EH) {
    using namespace cfg;
    int idx = blockIdx.x * blockDim.x + threadIdx.x;
    if (idx >= EH) return;
    int e = idx >> 3, h = idx & (H - 1);
    int s, d; edge_sd(ei, e, E, s, d);
    float alpha = pedge[idx] / (denom[d * H + h] + 1e-16f);
    const float* sv = hp + (size_t)s * HID + h * C;
    float* dv = agg + (size_t)d * HID + h * C;
#pragma unroll
    for (int c = 0; c < C; ++c) atomicAdd(&dv[c], sv[c] * alpha);
}

// Clenshaw combine: out = 2*elu(agg + conv_bias) - second_last + alphas[aidx]*h0
__global__ void combine_kernel(const float* __restrict__ agg, const float* __restrict__ bias,
                               const float* __restrict__ second, const float* __restrict__ h0,
                               const float* __restrict__ alphas, int aidx,
                               float* __restrict__ out, int total) {
    using namespace cfg;
    int idx = blockIdx.x * blockDim.x + threadIdx.x;
    if (idx >= total) return;
    int col = idx & (HID - 1);
    float t = agg[idx] + bias[col];
    t = t > 0.f ? t : (expf(t) - 1.f);                 // ELU
    out[idx] = 2.f * t - second[idx] + alphas[aidx] * h0[idx];
}

// row-wise log_softmax over OUT_C=40
__global__ void logsoftmax_kernel(const float* __restrict__ in, float* __restrict__ out, int N) {
    using namespace cfg;
    int r = blockIdx.x * blockDim.x + threadIdx.x;
    if (r >= N) return;
    const float* v = in + (size_t)r * OUT_C;
    float m = -INFINITY;
#pragma unroll
    for (int c = 0; c < OUT_C; ++c) m = fmaxf(m, v[c]);
    float ssum = 0.f;
#pragma unroll
    for (int c = 0; c < OUT_C; ++c) ssum += expf(v[c] - m);
    float ls = logf(ssum) + m;
    float* o = out + (size_t)r * OUT_C;
#pragma unroll
    for (int c = 0; c < OUT_C; ++c) o[c] = v[c] - ls;
}

// ---------------------------------------------------------------------------
extern "C" void kernel_launch(void* const* d_in, const int* in_sizes, int n_in,
                              void* d_out, int out_size, void* d_ws, size_t ws_size,
                              hipStream_t stream) {
    using namespace cfg;
    const float* x       = (const float*)d_in[0];
    const int*   ei      = (const int*)  d_in[1];
    const float* fc0_w   = (const float*)d_in[2];
    const float* fc0_b   = (const float*)d_in[3];
    const float* lin_w   = (const float*)d_in[4];
    const float* att_src = (const float*)d_in[5];
    const float* att_dst = (const float*)d_in[6];
    const float* conv_b  = (const float*)d_in[7];
    const float* alphas  = (const float*)d_in[8];
    const float* fc1_w   = (const float*)d_in[9];
    const float* fc1_b   = (const float*)d_in[10];

    const int N  = in_sizes[0] / IN_C;     // 50000
    const int E  = in_sizes[1] / 2;        // 800000
    const int EH = (E + N) * H;            // edge-head pairs incl. self loops
    const bool mdiv16 = (N % 16) == 0;     // true for the reference shape

    float* ws = (float*)d_ws;
    size_t off = 0;
    auto alloc = [&](size_t n) { float* p = ws + off; off += (n + 15) & ~(size_t)15; return p; };
    float* h0    = alloc((size_t)N * HID);
    float* buf0  = alloc((size_t)N * HID);
    float* buf1  = alloc((size_t)N * HID);
    float* buf2  = alloc((size_t)N * HID);
    float* hp    = alloc((size_t)N * HID);
    float* agg   = alloc((size_t)N * HID);
    float* a_s   = alloc((size_t)N * H);
    float* a_d   = alloc((size_t)N * H);
    float* emax  = alloc((size_t)N * H);
    float* denom = alloc((size_t)N * H);
    float* pedge = alloc((size_t)EH);
    float* fc1o  = alloc((size_t)N * OUT_C);
    (void)ws_size; (void)n_in; (void)out_size;

    const int TB = 256;
    const int gNH   = (N * HID + TB - 1) / TB;
    const int gEH   = (EH + TB - 1) / TB;
    const int gTile = (N + 15) / 16;

    // last = second_last = 0
    zero2_kernel<<<gNH, TB, 0, stream>>>(buf0, buf1, N * HID);

    // h0 = x @ fc0_w + fc0_b   (WMMA, K=512, 4 waves x 2 tiles)
    if (mdiv16)
        gemm_wmma_f32<IN_C, HID, 2, false, true><<<gTile, 4 * 32, 0, stream>>>(x, fc0_w, fc0_b, h0, N);
    else
        gemm_wmma_f32<IN_C, HID, 2, true,  true><<<gTile, 4 * 32, 0, stream>>>(x, fc0_w, fc0_b, h0, N);

    float* last = buf0; float* second = buf1; float* nxt = buf2;
    for (int i = 0; i < L; ++i) {
        init_layer_kernel<<<gNH, TB, 0, stream>>>(agg, emax, denom, N * HID, N * H);
        // hp = last @ lin_w[i]   (WMMA, K=128)
        const float* Wl = lin_w + (size_t)i * HID * HID;
        if (mdiv16)
            gemm_wmma_f32<HID, HID, 2, false, false><<<gTile, 4 * 32, 0, stream>>>(last, Wl, nullptr, hp, N);
        else
            gemm_wmma_f32<HID, HID, 2, true,  false><<<gTile, 4 * 32, 0, stream>>>(last, Wl, nullptr, hp, N);
        att_kernel<<<(N * H + TB - 1) / TB, TB, 0, stream>>>(
            hp, att_src + i * H * C, att_dst + i * H * C, a_s, a_d, N * H);
        edge_max_kernel<<<gEH, TB, 0, stream>>>(ei, a_s, a_d, emax, E, EH);
        edge_exp_sum_kernel<<<gEH, TB, 0, stream>>>(ei, a_s, a_d, emax, denom, pedge, E, EH);
        edge_scatter_kernel<<<gEH, TB, 0, stream>>>(ei, hp, pedge, denom, agg, E, EH);
        combine_kernel<<<gNH, TB, 0, stream>>>(
            agg, conv_b + i * HID, second, h0, alphas, (L - 1 - i), nxt, N * HID);
        float* t = second; second = last; last = nxt; nxt = t;
    }

    // fc1 (WMMA, NCOL=40 -> 3 masked tiles) + log_softmax
    gemm_wmma_f32<HID, OUT_C, 1, true, true><<<gTile, 3 * 32, 0, stream>>>(last, fc1_w, fc1_b, fc1o, N);
    logsoftmax_kernel<<<(N + TB - 1) / TB, TB, 0, stream>>>(fc1o, (float*)d_out, N);
}